// GCN_10436770529875
// MI455X (gfx1250) — compile-verified
//
#include <hip/hip_runtime.h>
#include <hip/hip_bf16.h>

typedef __attribute__((ext_vector_type(2))) float v2f;
typedef __attribute__((ext_vector_type(8))) float v8f;

#define D_FEAT 128

// ---------------- utility kernels ----------------

__global__ void zero_f32(float* __restrict__ p, size_t n) {
    size_t i = (size_t)blockIdx.x * blockDim.x + threadIdx.x;
    if (i < n) p[i] = 0.0f;
}

// out_deg/in_deg accumulation with hardware f32 atomics
__global__ void degree_kernel(const int* __restrict__ src, const int* __restrict__ dst,
                              float* __restrict__ outdeg, float* __restrict__ indeg, int E) {
    int e = blockIdx.x * blockDim.x + threadIdx.x;
    if (e < E) {
        unsafeAtomicAdd(&outdeg[src[e]], 1.0f);
        unsafeAtomicAdd(&indeg[dst[e]], 1.0f);
    }
}

// deg -> clamp(deg,1)^-0.5, applied in place over both norm arrays at once
__global__ void norm_kernel(float* __restrict__ p, int n) {
    int i = blockIdx.x * blockDim.x + threadIdx.x;
    if (i < n) {
        float d = fmaxf(p[i], 1.0f);
        p[i] = rsqrtf(d);
    }
}

// msg = (h * norm_src)[src] scatter-summed into agg[dst]; 32 lanes x float4 per edge
__global__ void scatter_kernel(const float* __restrict__ h, const float* __restrict__ norm_src,
                               const int* __restrict__ src, const int* __restrict__ dst,
                               float* __restrict__ agg, int E) {
    int t = blockIdx.x * blockDim.x + threadIdx.x;
    int e = t >> 5;               // 32 float4 chunks per 128-wide row
    int c = (t & 31) << 2;
    if (e < E) {
        int s = src[e];
        int d = dst[e];
        float ns = norm_src[s];
        float4 v = *(const float4*)(h + (size_t)s * D_FEAT + c);
        float* ap = agg + (size_t)d * D_FEAT + c;
        unsafeAtomicAdd(ap + 0, v.x * ns);
        unsafeAtomicAdd(ap + 1, v.y * ns);
        unsafeAtomicAdd(ap + 2, v.z * ns);
        unsafeAtomicAdd(ap + 3, v.w * ns);
    }
}

// SumPooling: hg[graph_ids[n]] += h[n]
__global__ void pool_kernel(const float* __restrict__ h, const int* __restrict__ gid,
                            float* __restrict__ hg, int N) {
    int t = blockIdx.x * blockDim.x + threadIdx.x;
    int n = t >> 5;
    int c = (t & 31) << 2;
    if (n < N) {
        int g = gid[n];
        float4 v = *(const float4*)(h + (size_t)n * D_FEAT + c);
        float* hp = hg + (size_t)g * D_FEAT + c;
        unsafeAtomicAdd(hp + 0, v.x);
        unsafeAtomicAdd(hp + 1, v.y);
        unsafeAtomicAdd(hp + 2, v.z);
        unsafeAtomicAdd(hp + 3, v.w);
    }
}

// ---------------- fp32 WMMA GEMM, 64x16 per wave (4 M-tiles register-blocked) --------
// One wave computes a 64x16 output tile as 4 stacked 16x16 WMMA tiles, reusing each
// B fragment for 4 v_wmma_f32_16x16x4_f32 issues (4x arithmetic intensity on B).
//   mode 0: out = c                      (plain)
//   mode 1: out = relu(resid + relu?(c)) (GCN layer; A rows pre-scaled by row_scale)
//   mode 2: out = relu(c)                (MLP hidden layers)
// A is Mrows x K (row-major), B is K x Ncol (row-major). K % 4 == 0, Ncol % 16 == 0.
// Row loads clamp to Mrows-1 so EXEC stays all-ones through every WMMA (ISA
// requirement); stores are masked per element after the last WMMA.
#define MT 4   // M-tiles per wave
__global__ void __launch_bounds__(32)
gemm_wmma_f32(const float* __restrict__ A, const float* __restrict__ B,
              const float* __restrict__ bias,
              const float* __restrict__ row_scale,
              const float* __restrict__ resid,
              float* __restrict__ out,
              int Mrows, int K, int Ncol, int relu_c, int mode) {
    int tm   = blockIdx.x;        // 64-row tile index
    int tn   = blockIdx.y;        // 16-col tile index
    int lane = threadIdx.x;
    int half = lane >> 4;         // 0: K=k0,k0+1  1: K=k0+2,k0+3  (f32 fragment layout)
    int l15  = lane & 15;

    int col = tn * 16 + l15;
    float bv = bias[col];

    const float* arow[MT];
    float rs[MT];
#pragma unroll
    for (int i = 0; i < MT; ++i) {
        int row = tm * (16 * MT) + i * 16 + l15;
        if (row >= Mrows) row = Mrows - 1;          // clamp: keeps EXEC full
        arow[i] = A + (size_t)row * K;
        rs[i] = row_scale ? row_scale[row] : 1.0f;
    }

    v8f acc[MT] = {};
    for (int k0 = 0; k0 < K; k0 += 4) {
        int ka = k0 + half * 2;
        v2f b;
        b.x = B[(size_t)ka * Ncol + col];
        b.y = B[(size_t)(ka + 1) * Ncol + col];
#pragma unroll
        for (int i = 0; i < MT; ++i) {
            __builtin_prefetch(arow[i] + k0 + 64, 0, 0);   // global_prefetch_b8
            v2f a = *(const v2f*)(arow[i] + ka);           // global_load_b64
            a.x *= rs[i];
            a.y *= rs[i];
            acc[i] = __builtin_amdgcn_wmma_f32_16x16x4_f32(
                /*neg_a=*/false, a, /*neg_b=*/false, b,
                /*c_mod=*/(short)0, acc[i], /*reuse_a=*/false, /*reuse_b=*/false);
        }
    }

#pragma unroll
    for (int i = 0; i < MT; ++i) {
#pragma unroll
        for (int v = 0; v < 8; ++v) {
            // C/D layout: lanes 0-15 -> M=v, lanes 16-31 -> M=v+8
            int m = tm * (16 * MT) + i * 16 + half * 8 + v;
            if (m < Mrows) {
                float c = acc[i][v] + bv;
                if (relu_c) c = fmaxf(c, 0.0f);
                float o;
                if (mode == 1)      o = fmaxf(resid[(size_t)m * Ncol + col] + c, 0.0f);
                else if (mode == 2) o = fmaxf(c, 0.0f);
                else                o = c;
                out[(size_t)m * Ncol + col] = o;
            }
        }
    }
}

// ---------------- host orchestration ----------------

extern "C" void kernel_launch(void* const* d_in, const int* in_sizes, int n_in,
                              void* d_out, int out_size, void* d_ws, size_t ws_size,
                              hipStream_t stream) {
    const float* node_features = (const float*)d_in[0];
    const int*   edge_src      = (const int*)d_in[1];
    const int*   edge_dst      = (const int*)d_in[2];
    const int*   graph_ids     = (const int*)d_in[3];
    const float* W_proj        = (const float*)d_in[4];
    const float* b_proj        = (const float*)d_in[5];
    const float* W_gcn         = (const float*)d_in[6];
    const float* b_gcn         = (const float*)d_in[7];
    const float* W_mlp0        = (const float*)d_in[8];
    const float* b_mlp0        = (const float*)d_in[9];
    const float* W_mlp1        = (const float*)d_in[10];
    const float* b_mlp1        = (const float*)d_in[11];
    const float* W_mlp2        = (const float*)d_in[12];
    const float* b_mlp2        = (const float*)d_in[13];

    const int D = D_FEAT;          // 128
    const int Mm = 256;            // mlp_dim
    const int N = in_sizes[3];     // graph_ids: [N]
    const int E = in_sizes[1];     // edges
    const int L = in_sizes[6] / (D * D);
    const int G = out_size / Mm;   // graphs

    float* ws = (float*)d_ws;
    float* norm_src = ws;                         // N
    float* norm_dst = norm_src + N;               // N
    float* h        = norm_dst + N;               // N*D
    float* agg      = h + (size_t)N * D;          // N*D
    float* hg       = agg + (size_t)N * D;        // G*D
    float* t0       = hg + (size_t)G * D;         // G*Mm
    float* t1       = t0 + (size_t)G * Mm;        // G*Mm

    const int TB = 256;
    auto blocks = [&](size_t n) { return (unsigned)((n + TB - 1) / TB); };
    const int MROWS = 16 * MT;     // 64 rows per wave-tile

    // 1) degree -> symmetric norms
    zero_f32<<<blocks((size_t)2 * N), TB, 0, stream>>>(norm_src, (size_t)2 * N);
    degree_kernel<<<blocks(E), TB, 0, stream>>>(edge_src, edge_dst, norm_src, norm_dst, E);
    norm_kernel<<<blocks((size_t)2 * N), TB, 0, stream>>>(norm_src, 2 * N);

    // 2) h = node_features @ W_proj + b_proj
    {
        dim3 g((N + MROWS - 1) / MROWS, D / 16);
        gemm_wmma_f32<<<g, 32, 0, stream>>>(node_features, W_proj, b_proj,
                                            nullptr, nullptr, h, N, D, D,
                                            /*relu_c=*/0, /*mode=*/0);
    }

    // 3) L GraphConv layers with residual + ReLU
    for (int i = 0; i < L; ++i) {
        zero_f32<<<blocks((size_t)N * D), TB, 0, stream>>>(agg, (size_t)N * D);
        scatter_kernel<<<blocks((size_t)E * 32), TB, 0, stream>>>(h, norm_src,
                                                                  edge_src, edge_dst, agg, E);
        dim3 g((N + MROWS - 1) / MROWS, D / 16);
        // c = (agg * norm_dst) @ W_gcn[i] + b_gcn[i]; relu on c except last layer;
        // h = relu(h + c)   (in place: each element read once in epilogue, then written)
        gemm_wmma_f32<<<g, 32, 0, stream>>>(agg, W_gcn + (size_t)i * D * D,
                                            b_gcn + (size_t)i * D,
                                            norm_dst, h, h, N, D, D,
                                            /*relu_c=*/(i < L - 1) ? 1 : 0, /*mode=*/1);
    }

    // 4) SumPooling -> hg [G, D]
    zero_f32<<<blocks((size_t)G * D), TB, 0, stream>>>(hg, (size_t)G * D);
    pool_kernel<<<blocks((size_t)N * 32), TB, 0, stream>>>(h, graph_ids, hg, N);

    // 5) MLP: t0 = relu(hg@W0+b0); t1 = relu(t0@W1+b1); out = t1@W2+b2
    {
        dim3 g0((G + MROWS - 1) / MROWS, Mm / 16);
        gemm_wmma_f32<<<g0, 32, 0, stream>>>(hg, W_mlp0, b_mlp0, nullptr, nullptr,
                                             t0, G, D, Mm, 0, /*mode=*/2);
        gemm_wmma_f32<<<g0, 32, 0, stream>>>(t0, W_mlp1, b_mlp1, nullptr, nullptr,
                                             t1, G, Mm, Mm, 0, /*mode=*/2);
        gemm_wmma_f32<<<g0, 32, 0, stream>>>(t1, W_mlp2, b_mlp2, nullptr, nullptr,
                                             (float*)d_out, G, Mm, Mm, 0, /*mode=*/0);
    }
}